// TemporalContrastiveLoss_18124761989925
// MI455X (gfx1250) — compile-verified
//
#include <hip/hip_runtime.h>
#include <cmath>

// CDNA5 wave32 WMMA operand types
typedef __attribute__((ext_vector_type(2))) float v2f;  // A/B for V_WMMA_F32_16X16X4_F32 (16x4 / 4x16 fp32)
typedef __attribute__((ext_vector_type(8))) float v8f;  // 16x16 fp32 C/D accumulator

#define NTHREADS 256
#define NBLOCKS  2048

__device__ __forceinline__ float fold8(v8f a) {
  // Sum of this lane's 8 accumulator rows. Summed over all 32 lanes of a wave
  // this equals 16x the true wave total (ones-B replicates row sums across the
  // 16 columns), corrected by an exact *0.0625f at the caller.
  return ((a[0] + a[1]) + (a[2] + a[3])) + ((a[4] + a[5]) + (a[6] + a[7]));
}

// ---------------------------------------------------------------------------
// Pass 1: per-block partial {sum, sumsq} over all N elements.
// WMMA-with-ones as the accumulation engine (full fp32, EXEC all-ones region).
// ---------------------------------------------------------------------------
__global__ __launch_bounds__(NTHREADS)
void tcl_stats_pass(const float4* __restrict__ x, float* __restrict__ part, long nvec4) {
  const v2f ones = {1.0f, 1.0f};
  v8f accS = {};
  v8f accQ = {};
  float exS = 0.0f, exQ = 0.0f;

  const long tid    = (long)blockIdx.x * blockDim.x + threadIdx.x;
  const long stride = (long)gridDim.x * blockDim.x;
  const long iters  = nvec4 / stride;  // wave-uniform trip count -> EXEC==all-ones for WMMA

  for (long k = 0; k < iters; ++k) {
    float4 v = x[tid + k * stride];
    v2f a; a.x = v.x + v.y;                    a.y = v.z + v.w;
    v2f q; q.x = fmaf(v.x, v.x, v.y * v.y);    q.y = fmaf(v.z, v.z, v.w * v.w);
    accS = __builtin_amdgcn_wmma_f32_16x16x4_f32(false, a, false, ones, (short)0, accS, false, false);
    accQ = __builtin_amdgcn_wmma_f32_16x16x4_f32(false, q, false, ones, (short)0, accQ, false, false);
  }
  // ragged tail (divergent EXEC possible): plain VALU, no WMMA
  for (long i = iters * stride + tid; i < nvec4; i += stride) {
    float4 v = x[i];
    exS += (v.x + v.y) + (v.z + v.w);
    exQ += fmaf(v.x, v.x, v.y * v.y) + fmaf(v.z, v.z, v.w * v.w);
  }

  __shared__ float ls[NTHREADS];
  __shared__ float lq[NTHREADS];
  ls[threadIdx.x] = fold8(accS) * 0.0625f + exS;
  lq[threadIdx.x] = fold8(accQ) * 0.0625f + exQ;
  __syncthreads();
  for (int off = NTHREADS / 2; off > 0; off >>= 1) {
    if ((int)threadIdx.x < off) {
      ls[threadIdx.x] += ls[threadIdx.x + off];
      lq[threadIdx.x] += lq[threadIdx.x + off];
    }
    __syncthreads();
  }
  if (threadIdx.x == 0) {
    part[2 * blockIdx.x + 0] = ls[0];
    part[2 * blockIdx.x + 1] = lq[0];
  }
}

// ---------------------------------------------------------------------------
// Pass 2: reduce block partials (double) -> mu and c = -0.5/var (ddof=1).
// ---------------------------------------------------------------------------
__global__ __launch_bounds__(NTHREADS)
void tcl_stats_finalize(const float* __restrict__ part, float* __restrict__ consts,
                        int nblk, double N) {
  __shared__ double ls[NTHREADS];
  __shared__ double lq[NTHREADS];
  double s = 0.0, q = 0.0;
  for (int i = threadIdx.x; i < nblk; i += NTHREADS) {
    s += (double)part[2 * i + 0];
    q += (double)part[2 * i + 1];
  }
  ls[threadIdx.x] = s;
  lq[threadIdx.x] = q;
  __syncthreads();
  for (int off = NTHREADS / 2; off > 0; off >>= 1) {
    if ((int)threadIdx.x < off) {
      ls[threadIdx.x] += ls[threadIdx.x + off];
      lq[threadIdx.x] += lq[threadIdx.x + off];
    }
    __syncthreads();
  }
  if (threadIdx.x == 0) {
    double mu  = ls[0] / N;
    double var = (lq[0] - ls[0] * ls[0] / N) / (N - 1.0);  // unbiased, like torch.std
    consts[0] = (float)mu;
    consts[1] = (float)(-0.5 / var);
  }
}

// ---------------------------------------------------------------------------
// Pass 3: per-block partial sum of gw = exp(c*(x-mu)^2). WMMA accumulation.
// x fits in the 192MB L2 after pass 1, so default (RT) temporal hints.
// ---------------------------------------------------------------------------
__global__ __launch_bounds__(NTHREADS)
void tcl_gauss_pass(const float4* __restrict__ x, const float* __restrict__ consts,
                    float* __restrict__ gpart, long nvec4) {
  const float mu = consts[0];
  const float c  = consts[1];
  const v2f ones = {1.0f, 1.0f};
  v8f acc = {};
  float ex = 0.0f;

  const long tid    = (long)blockIdx.x * blockDim.x + threadIdx.x;
  const long stride = (long)gridDim.x * blockDim.x;
  const long iters  = nvec4 / stride;

  for (long k = 0; k < iters; ++k) {
    float4 v = x[tid + k * stride];
    float d0 = v.x - mu, d1 = v.y - mu, d2 = v.z - mu, d3 = v.w - mu;
    float e0 = expf(c * d0 * d0);
    float e1 = expf(c * d1 * d1);
    float e2 = expf(c * d2 * d2);
    float e3 = expf(c * d3 * d3);
    v2f a; a.x = e0 + e1; a.y = e2 + e3;
    acc = __builtin_amdgcn_wmma_f32_16x16x4_f32(false, a, false, ones, (short)0, acc, false, false);
  }
  for (long i = iters * stride + tid; i < nvec4; i += stride) {
    float4 v = x[i];
    float d0 = v.x - mu, d1 = v.y - mu, d2 = v.z - mu, d3 = v.w - mu;
    ex += (expf(c * d0 * d0) + expf(c * d1 * d1)) + (expf(c * d2 * d2) + expf(c * d3 * d3));
  }

  __shared__ float ls[NTHREADS];
  ls[threadIdx.x] = fold8(acc) * 0.0625f + ex;
  __syncthreads();
  for (int off = NTHREADS / 2; off > 0; off >>= 1) {
    if ((int)threadIdx.x < off) ls[threadIdx.x] += ls[threadIdx.x + off];
    __syncthreads();
  }
  if (threadIdx.x == 0) gpart[blockIdx.x] = ls[0];
}

// ---------------------------------------------------------------------------
// Pass 4: S = sum(gpart); row i=T-1: loss = (T-1)*sum_j w^2*log(w), w = g_j/S;
// out = loss + log1p(exp(1e4*loss)).
// ---------------------------------------------------------------------------
__global__ __launch_bounds__(NTHREADS)
void tcl_finalize(const float* __restrict__ x, const float* __restrict__ consts,
                  const float* __restrict__ gpart, int nblk, int T,
                  float* __restrict__ out) {
  __shared__ double ls[NTHREADS];
  __shared__ double Ssum;

  double s = 0.0;
  for (int i = threadIdx.x; i < nblk; i += NTHREADS) s += (double)gpart[i];
  ls[threadIdx.x] = s;
  __syncthreads();
  for (int off = NTHREADS / 2; off > 0; off >>= 1) {
    if ((int)threadIdx.x < off) ls[threadIdx.x] += ls[threadIdx.x + off];
    __syncthreads();
  }
  if (threadIdx.x == 0) Ssum = ls[0];
  __syncthreads();
  const double S  = Ssum;
  const float  mu = consts[0];
  const float  c  = consts[1];

  const float* __restrict__ row = x + (size_t)(T - 1) * (size_t)T;
  double acc = 0.0;
  for (int j = threadIdx.x; j < T; j += NTHREADS) {
    float  d = row[j] - mu;
    double w = (double)expf(c * d * d) / S;
    acc += w * w * log(w);
  }
  ls[threadIdx.x] = acc;
  __syncthreads();
  for (int off = NTHREADS / 2; off > 0; off >>= 1) {
    if ((int)threadIdx.x < off) ls[threadIdx.x] += ls[threadIdx.x + off];
    __syncthreads();
  }
  if (threadIdx.x == 0) {
    double loss = (double)(T - 1) * ls[0];
    out[0] = (float)(loss + log1p(exp(10000.0 * loss)));
  }
}

extern "C" void kernel_launch(void* const* d_in, const int* in_sizes, int n_in,
                              void* d_out, int out_size, void* d_ws, size_t ws_size,
                              hipStream_t stream) {
  (void)n_in; (void)out_size; (void)ws_size;
  const float* x = (const float*)d_in[0];
  const long   N = (long)in_sizes[0];                 // T*T
  const int    T = (int)llround(std::sqrt((double)N));
  const long   nvec4 = N / 4;                         // T=4096 -> divisible

  // Workspace layout (floats): [0,2*NBLOCKS) stats partials | +4 consts | gauss partials
  float* ws     = (float*)d_ws;
  float* part   = ws;
  float* consts = ws + 2 * NBLOCKS;
  float* gpart  = ws + 2 * NBLOCKS + 4;

  tcl_stats_pass    <<<NBLOCKS, NTHREADS, 0, stream>>>((const float4*)x, part, nvec4);
  tcl_stats_finalize<<<1,       NTHREADS, 0, stream>>>(part, consts, NBLOCKS, (double)N);
  tcl_gauss_pass    <<<NBLOCKS, NTHREADS, 0, stream>>>((const float4*)x, consts, gpart, nvec4);
  tcl_finalize      <<<1,       NTHREADS, 0, stream>>>(x, consts, gpart, NBLOCKS, T, (float*)d_out);
}